// MaxLen2_Pool_19902878449864
// MI455X (gfx1250) — compile-verified
//
#include <hip/hip_runtime.h>

// Guided length-2 max pool along W (W = 128, fp32), MI455X / gfx1250.
//
// out[..., w] = max(x[..., w], guide[..., w] * x[..., w-1]),  out[..., 0] = x[..., 0]
//
// Pure HBM-streaming kernel: 12 B/elem, ~402 MB total -> ~17 us floor at 23.3 TB/s.
// One wave32 per row (128 floats = 32 lanes x float4). Cross-lane left-neighbor
// via wave32 shuffle; b128 NT loads/stores for max bandwidth and L2 bypass.

typedef __attribute__((ext_vector_type(4))) float v4f;

__global__ __launch_bounds__(256)
void guided_maxpool_w128(const float* __restrict__ x,
                         const float* __restrict__ g,
                         float* __restrict__ out,
                         int nrows) {
    const int lane = threadIdx.x & 31;          // lane within wave32
    const int wave = threadIdx.x >> 5;          // wave within block (8 waves)
    const int row  = blockIdx.x * 8 + wave;     // one row of 128 floats per wave
    if (row >= nrows) return;

    const long long base = (long long)row * 128 + (lane << 2);

    // Streaming loads: non-temporal (th:NT) -> don't pollute L2 (384 MB > 192 MB L2).
    v4f xv = __builtin_nontemporal_load((const v4f*)(x + base));
    v4f gv = __builtin_nontemporal_load((const v4f*)(g + base));

    // Left neighbor of this lane's first element = previous lane's last element.
    // wave32 shuffle: no LDS storage, no extra global traffic.
    float prev = __shfl_up(xv.w, 1, 32);

    v4f o;
    // lane 0 holds w = 0: no left neighbor, out = x (reference masks cand to -inf).
    o.x = (lane == 0) ? xv.x : fmaxf(xv.x, gv.x * prev);
    o.y = fmaxf(xv.y, gv.y * xv.x);
    o.z = fmaxf(xv.z, gv.z * xv.y);
    o.w = fmaxf(xv.w, gv.w * xv.z);

    __builtin_nontemporal_store(o, (v4f*)(out + base));
}

extern "C" void kernel_launch(void* const* d_in, const int* in_sizes, int n_in,
                              void* d_out, int out_size, void* d_ws, size_t ws_size,
                              hipStream_t stream) {
    const float* x = (const float*)d_in[0];
    const float* g = (const float*)d_in[1];
    float* o = (float*)d_out;

    const int total = in_sizes[0];      // B*C*H*W = 33,554,432
    const int nrows = total >> 7;       // W = 128 (fixed by the reference)

    const int rows_per_block = 8;       // 8 wave32 per 256-thread block
    const int blocks = (nrows + rows_per_block - 1) / rows_per_block;

    hipLaunchKernelGGL(guided_maxpool_w128,
                       dim3(blocks), dim3(256), 0, stream,
                       x, g, o, nrows);
}